// Attention_34608846471478
// MI455X (gfx1250) — compile-verified
//
#include <hip/hip_runtime.h>
#include <hip/hip_bf16.h>

typedef __attribute__((ext_vector_type(2))) float v2f;
typedef __attribute__((ext_vector_type(8))) float v8f;

__device__ __forceinline__ v8f wmma4(v2f a, v2f b, v8f c) {
  // 8-arg pattern (probe-confirmed arity for _16x16x4):
  // (neg_a, A, neg_b, B, c_mod, C, reuse_a, reuse_b)
  return __builtin_amdgcn_wmma_f32_16x16x4_f32(
      false, a, false, b, (short)0, c, false, false);
}

// ---------------------------------------------------------------------------
// Kernel 1: QKV projection.
// X rows: m = bp*1024 + s, bp = b*2 + p; p=0 -> normalized_resid_pre, p=1 -> alt.
// For (m_tile, head n, sel in {Q,K,V}) one wave computes a 16x64 tile.
// Output layout: [bp][n][s][h] (rows of 64 contiguous) for the attention kernel.
// ---------------------------------------------------------------------------
__global__ __launch_bounds__(32) void qkv_kernel(
    const float* __restrict__ x0, const float* __restrict__ x1,
    const float* __restrict__ WQ, const float* __restrict__ bQ,
    const float* __restrict__ WK, const float* __restrict__ bK,
    const float* __restrict__ WV, const float* __restrict__ bV,
    float* __restrict__ Qw, float* __restrict__ Kw, float* __restrict__ Vw) {
  const int mt  = blockIdx.x;          // 0..511
  const int n   = blockIdx.y;          // head 0..11
  const int sel = blockIdx.z;          // 0=Q 1=K 2=V
  const float* W    = (sel == 0) ? WQ : (sel == 1) ? WK : WV;
  const float* bias = (sel == 0) ? bQ : (sel == 1) ? bK : bV;
  float*       outp = (sel == 0) ? Qw : (sel == 1) ? Kw : Vw;

  const int lane = threadIdx.x;
  const int h2 = lane >> 4;            // half-wave select (K pair / row group)
  const int lr = lane & 15;

  const int m0 = mt * 16;
  const int bp = m0 >> 10;
  const int s0 = m0 & 1023;
  const float* xbase = (bp & 1) ? x1 : x0;
  const float* xrow  = xbase + ((size_t)((bp >> 1) * 1024 + s0 + lr)) * 768;

  const float* Wn = W + (size_t)n * 768 * 64;
  v8f acc[4] = {};
  for (int k = 0; k < 768; k += 4) {
    v2f a;
    a.x = xrow[k + 2 * h2];
    a.y = xrow[k + 2 * h2 + 1];
#pragma unroll
    for (int c = 0; c < 4; ++c) {
      v2f bv;
      bv.x = Wn[(size_t)(k + 2 * h2) * 64 + c * 16 + lr];
      bv.y = Wn[(size_t)(k + 2 * h2 + 1) * 64 + c * 16 + lr];
      acc[c] = wmma4(a, bv, acc[c]);
    }
  }

  float* obase = outp + (size_t)(bp * 12 + n) * 1024 * 64;
#pragma unroll
  for (int c = 0; c < 4; ++c) {
    const float bb = bias[n * 64 + c * 16 + lr];
#pragma unroll
    for (int r = 0; r < 8; ++r) {
      const int row = r + 8 * h2;
      obase[(size_t)(s0 + row) * 64 + c * 16 + lr] = acc[c][r] + bb;
    }
  }
}

// ---------------------------------------------------------------------------
// Kernel 2: causal flash attention per (bp, head, 16-query tile), one wave.
// S = Q K^T / 8 built from 16 chained V_WMMA_F32_16X16X4_F32; online softmax
// with 16-lane butterfly reductions (C-layout rows occupy 16 contiguous
// lanes); P transposed via LDS; Z kept as 4 fp32 16x16 accumulators.
// Z output layout: [p][b][q][n*64+h]  (so (p,b,q) rows are 768 contiguous).
// ---------------------------------------------------------------------------
__global__ __launch_bounds__(32) void attn_kernel(
    const float* __restrict__ Qw, const float* __restrict__ Kw,
    const float* __restrict__ Vw, float* __restrict__ Zw) {
  const int qt = blockIdx.x;           // 0..63 query tile
  const int n  = blockIdx.y;           // head
  const int bp = blockIdx.z;           // 0..7
  const int lane = threadIdx.x;
  const int h2 = lane >> 4;
  const int lr = lane & 15;
  const int m0 = qt * 16;

  const float* Qb = Qw + (size_t)(bp * 12 + n) * 1024 * 64;
  const float* Kb = Kw + (size_t)(bp * 12 + n) * 1024 * 64;
  const float* Vb = Vw + (size_t)(bp * 12 + n) * 1024 * 64;

  // Preload Q tile as 16 A-fragments (covers channels 0..63).
  v2f qa[16];
  const float* qrow = Qb + (size_t)(m0 + lr) * 64;
#pragma unroll
  for (int t = 0; t < 16; ++t) {
    qa[t].x = qrow[4 * t + 2 * h2];
    qa[t].y = qrow[4 * t + 2 * h2 + 1];
  }

  v8f zc[4] = {};
  float mstate[8], lstate[8];
#pragma unroll
  for (int r = 0; r < 8; ++r) { mstate[r] = -1e30f; lstate[r] = 0.0f; }

  __shared__ float pt[256];            // 16x16 P transpose staging

  for (int jb = 0; jb <= qt; ++jb) {
    const int j0 = jb * 16;
    // S tile = Q (16x64) * K^T (64x16)
    v8f sacc = {};
    const float* krow = Kb + (size_t)(j0 + lr) * 64;
#pragma unroll
    for (int t = 0; t < 16; ++t) {
      v2f kb;
      kb.x = krow[4 * t + 2 * h2];
      kb.y = krow[4 * t + 2 * h2 + 1];
      sacc = wmma4(qa[t], kb, sacc);
    }

    float sv[8];
#pragma unroll
    for (int r = 0; r < 8; ++r) {
      float v = sacc[r] * 0.125f;      // 1/sqrt(64)
      if (jb == qt) {                  // causal mask on diagonal block only
        const int qpos = m0 + r + 8 * h2;
        const int kpos = j0 + lr;
        if (kpos > qpos) v = -1e30f;
      }
      sv[r] = v;
    }

    // Online softmax update, per row (row = r + 8*h2 lives in this lane's
    // 16-lane group, so butterfly reductions stay inside the group).
#pragma unroll
    for (int r = 0; r < 8; ++r) {
      float rm = sv[r];
      rm = fmaxf(rm, __shfl_xor(rm, 1));
      rm = fmaxf(rm, __shfl_xor(rm, 2));
      rm = fmaxf(rm, __shfl_xor(rm, 4));
      rm = fmaxf(rm, __shfl_xor(rm, 8));
      const float newm  = fmaxf(mstate[r], rm);
      const float alpha = __expf(mstate[r] - newm);
      const float p     = __expf(sv[r] - newm);
      float rs = p;
      rs += __shfl_xor(rs, 1);
      rs += __shfl_xor(rs, 2);
      rs += __shfl_xor(rs, 4);
      rs += __shfl_xor(rs, 8);
      lstate[r] = lstate[r] * alpha + rs;
      mstate[r] = newm;
      sv[r] = p;
#pragma unroll
      for (int c = 0; c < 4; ++c) zc[c][r] *= alpha;
    }

    // Transpose P (C layout -> A layout) through LDS.
    __syncthreads();
#pragma unroll
    for (int r = 0; r < 8; ++r) pt[(r + 8 * h2) * 16 + lr] = sv[r];
    __syncthreads();
    v2f pa[4];
#pragma unroll
    for (int t = 0; t < 4; ++t) {
      pa[t].x = pt[lr * 16 + 4 * t + 2 * h2];
      pa[t].y = pt[lr * 16 + 4 * t + 2 * h2 + 1];
    }

    // Z += P (16x16) * V_block (16x64)
#pragma unroll
    for (int t = 0; t < 4; ++t) {
#pragma unroll
      for (int c = 0; c < 4; ++c) {
        v2f vb;
        vb.x = Vb[(size_t)(j0 + 4 * t + 2 * h2) * 64 + c * 16 + lr];
        vb.y = Vb[(size_t)(j0 + 4 * t + 2 * h2 + 1) * 64 + c * 16 + lr];
        zc[c] = wmma4(pa[t], vb, zc[c]);
      }
    }
  }

  // Normalize and store Z[p][b][q][n*64+h].
  const int p = bp & 1;
  const int b = bp >> 1;
  float* zbase = Zw + ((size_t)((p * 4 + b) * 1024 + m0)) * 768 + n * 64;
#pragma unroll
  for (int r = 0; r < 8; ++r) {
    const float inv = 1.0f / lstate[r];
#pragma unroll
    for (int c = 0; c < 4; ++c) {
      zbase[(size_t)(r + 8 * h2) * 768 + c * 16 + lr] = zc[c][r] * inv;
    }
  }
}

// ---------------------------------------------------------------------------
// Kernel 3: out[b, p, q, :] = Z_p[b,q,:] (768) @ W_O (768x768 row-major) + b_O
// M = 8192 rows over (p,b,q). One wave -> 16x64 tile.
// ---------------------------------------------------------------------------
__global__ __launch_bounds__(32) void outproj_kernel(
    const float* __restrict__ Zw, const float* __restrict__ WO,
    const float* __restrict__ bO, float* __restrict__ out) {
  const int mt = blockIdx.x;           // 0..511
  const int nt = blockIdx.y;           // 0..11 (column group of 64)
  const int lane = threadIdx.x;
  const int h2 = lane >> 4;
  const int lr = lane & 15;
  const int m0 = mt * 16;
  const int d0 = nt * 64;

  const float* arow = Zw + (size_t)(m0 + lr) * 768;
  v8f acc[4] = {};
  for (int k = 0; k < 768; k += 4) {
    v2f a;
    a.x = arow[k + 2 * h2];
    a.y = arow[k + 2 * h2 + 1];
#pragma unroll
    for (int c = 0; c < 4; ++c) {
      v2f bv;
      bv.x = WO[(size_t)(k + 2 * h2) * 768 + d0 + c * 16 + lr];
      bv.y = WO[(size_t)(k + 2 * h2 + 1) * 768 + d0 + c * 16 + lr];
      acc[c] = wmma4(a, bv, acc[c]);
    }
  }

#pragma unroll
  for (int r = 0; r < 8; ++r) {
    const int m = m0 + r + 8 * h2;
    const int p = m >> 12;
    const int b = (m >> 10) & 3;
    const int q = m & 1023;
    float* orow = out + ((size_t)((b * 14 + p) * 1024 + q)) * 768 + d0;
#pragma unroll
    for (int c = 0; c < 4; ++c) {
      orow[c * 16 + lr] = acc[c][r] + bO[d0 + c * 16 + lr];
    }
  }
}

// ---------------------------------------------------------------------------
// Kernel 4: out[b, 2+k, q, :] = out[b,1,q,:] + (Z0 - Z1)[b,q,k,:] @ W_O[k]
// Per head k: (4096 x 64) @ (64 x 768). One wave -> 16x64 tile.
// ---------------------------------------------------------------------------
__global__ __launch_bounds__(32) void ablate_kernel(
    const float* __restrict__ Zw, const float* __restrict__ WO,
    float* __restrict__ out) {
  const int mt = blockIdx.x;           // 0..255 over (b,q)/16
  const int nt = blockIdx.y;           // 0..11
  const int hd = blockIdx.z;           // head k
  const int lane = threadIdx.x;
  const int h2 = lane >> 4;
  const int lr = lane & 15;
  const int m0 = mt * 16;
  const int d0 = nt * 64;

  const float* z0 = Zw + (size_t)(m0 + lr) * 768 + hd * 64;          // p=0
  const float* z1 = Zw + (size_t)(4096 + m0 + lr) * 768 + hd * 64;   // p=1

  v8f acc[4] = {};
#pragma unroll
  for (int k = 0; k < 64; k += 4) {
    v2f a;
    a.x = z0[k + 2 * h2] - z1[k + 2 * h2];
    a.y = z0[k + 2 * h2 + 1] - z1[k + 2 * h2 + 1];
#pragma unroll
    for (int c = 0; c < 4; ++c) {
      v2f bv;
      bv.x = WO[(size_t)(hd * 64 + k + 2 * h2) * 768 + d0 + c * 16 + lr];
      bv.y = WO[(size_t)(hd * 64 + k + 2 * h2 + 1) * 768 + d0 + c * 16 + lr];
      acc[c] = wmma4(a, bv, acc[c]);
    }
  }

#pragma unroll
  for (int r = 0; r < 8; ++r) {
    const int m = m0 + r + 8 * h2;
    const int b = m >> 10;
    const int q = m & 1023;
    const float* t1 = out + ((size_t)((b * 14 + 1) * 1024 + q)) * 768 + d0;
    float* orow = out + ((size_t)((b * 14 + 2 + hd) * 1024 + q)) * 768 + d0;
#pragma unroll
    for (int c = 0; c < 4; ++c) {
      orow[c * 16 + lr] = acc[c][r] + t1[c * 16 + lr];
    }
  }
}

extern "C" void kernel_launch(void* const* d_in, const int* in_sizes, int n_in,
                              void* d_out, int out_size, void* d_ws, size_t ws_size,
                              hipStream_t stream) {
  const float* x0 = (const float*)d_in[0];  // normalized_resid_pre (4,1,1024,768)
  const float* x1 = (const float*)d_in[1];  // alt (4,1024,768)
  const float* WQ = (const float*)d_in[2];
  const float* bQ = (const float*)d_in[3];
  const float* WK = (const float*)d_in[4];
  const float* bK = (const float*)d_in[5];
  const float* WV = (const float*)d_in[6];
  const float* bV = (const float*)d_in[7];
  const float* WO = (const float*)d_in[8];
  const float* bO = (const float*)d_in[9];
  float* out = (float*)d_out;

  // Workspace partition (floats): Q/K/V each 8*12*1024*64, Z 2*4*1024*768.
  const size_t QKV_ELTS = (size_t)8 * 12 * 1024 * 64;   // 6,291,456
  float* Qw = (float*)d_ws;
  float* Kw = Qw + QKV_ELTS;
  float* Vw = Kw + QKV_ELTS;
  float* Zw = Vw + QKV_ELTS;

  qkv_kernel<<<dim3(512, 12, 3), 32, 0, stream>>>(
      x0, x1, WQ, bQ, WK, bK, WV, bV, Qw, Kw, Vw);
  attn_kernel<<<dim3(64, 12, 8), 32, 0, stream>>>(Qw, Kw, Vw, Zw);
  outproj_kernel<<<dim3(512, 12), 32, 0, stream>>>(Zw, WO, bO, out);
  ablate_kernel<<<dim3(256, 12, 12), 32, 0, stream>>>(Zw, WO, out);
}